// QuantumKernelMethod_65481071405355
// MI455X (gfx1250) — compile-verified
//
#include <hip/hip_runtime.h>
#include <hip/hip_bf16.h>
#include <math.h>

typedef __attribute__((ext_vector_type(2))) float v2f;
typedef __attribute__((ext_vector_type(8))) float v8f;

// ---------------------------------------------------------------------------
// Prologue: per-row scalars.
//   c[i]  = 0.5*||x_i - y_i||^2 - ||x_i||^2        (diag-normalization + xn fold)
//   q2[i] = 0.5*(cos^2(x[i,0]) + sin^2(x[i,0])*sin(x[i,1]))
//   yn[j] = ||y_j||^2
// ---------------------------------------------------------------------------
__global__ __launch_bounds__(256) void qkm_precompute(
    const float* __restrict__ x, const float* __restrict__ y,
    float* __restrict__ crow, float* __restrict__ q2, float* __restrict__ yn,
    int M, int N, int D)
{
    int i = blockIdx.x * blockDim.x + threadIdx.x;
    if (i < M) {
        const float* xr = x + (size_t)i * D;
        const float* yr = y + (size_t)i * D;
        float xn = 0.f, dd = 0.f;
        #pragma unroll 8
        for (int k = 0; k < D; ++k) {
            float xv = xr[k], yv = yr[k];
            xn += xv * xv;
            float t = xv - yv;
            dd += t * t;
        }
        crow[i] = 0.5f * dd - xn;
        float t0 = xr[0], t1 = xr[1];
        float c0 = cosf(t0), s0 = sinf(t0), s1 = sinf(t1);
        q2[i] = 0.5f * (c0 * c0 + s0 * s0 * s1);
    }
    if (i < N) {
        const float* yr = y + (size_t)i * D;
        float s = 0.f;
        #pragma unroll 8
        for (int k = 0; k < D; ++k) { float yv = yr[k]; s += yv * yv; }
        yn[i] = s;
    }
}

// ---------------------------------------------------------------------------
// Fused tile kernel: one wave32 per 16x16 output tile.
// Dot products via V_WMMA_F32_16X16X4_F32 (full f32 precision; exponents reach
// ~|100| so f16/bf8 inputs would turn dot error into % -level output error).
//
// f32 16x16x4 operand layout (ISA 7.12.2): lane L, VGPR v holds K = v + 2*(L/16);
// A rows / B cols indexed by L%16 -> both operands are a float2 load at
// [l16][4*kk + 2*half].  D: VGPR r -> row r + 8*half, col l16.
// ---------------------------------------------------------------------------
__global__ __launch_bounds__(256) void qkm_fused(
    const float* __restrict__ x, const float* __restrict__ y,
    const float* __restrict__ crow, const float* __restrict__ q2,
    const float* __restrict__ yn, float* __restrict__ out, int N)
{
    const int lane = threadIdx.x & 31;
    const int wave = threadIdx.x >> 5;          // 8 waves / block
    const int half = lane >> 4;                  // 0 or 1
    const int l16  = lane & 15;

    const int tile_col = blockIdx.x;             // 0..N/16-1
    const int tile_row = blockIdx.y * 8 + wave;  // 0..M/16-1
    const int row_base = tile_row * 16;
    const int col_base = tile_col * 16;

    const float* xrow = x + (size_t)(row_base + l16) * 64;
    const float* ycol = y + (size_t)(col_base + l16) * 64;

    v8f acc = {};
    #pragma unroll
    for (int kk = 0; kk < 16; ++kk) {
        const int k = kk * 4 + half * 2;         // K = v + 2*half within step
        v2f a = *(const v2f*)(xrow + k);         // A[m=l16][k..k+1]
        v2f b = *(const v2f*)(ycol + k);         // B[k..k+1][n=l16] = y[n][k..k+1]
        acc = __builtin_amdgcn_wmma_f32_16x16x4_f32(
            /*neg_a=*/false, a, /*neg_b=*/false, b,
            /*c_mod=*/(short)0, acc, /*reuse_a=*/false, /*reuse_b=*/false);
    }

    const float ynv = yn[col_base + l16];
    #pragma unroll
    for (int r = 0; r < 8; ++r) {
        const int row = row_base + r + half * 8;
        const float e = 2.0f * acc[r] + crow[row] - ynv;
        out[(size_t)row * N + col_base + l16] = q2[row] + 0.5f * __expf(e);
    }
}

// ---------------------------------------------------------------------------
extern "C" void kernel_launch(void* const* d_in, const int* in_sizes, int n_in,
                              void* d_out, int out_size, void* d_ws, size_t ws_size,
                              hipStream_t stream)
{
    const float* x = (const float*)d_in[0];
    const float* y = (const float*)d_in[1];
    float* out = (float*)d_out;

    const int D = 64;
    const int M = in_sizes[0] / D;   // 2048
    const int N = in_sizes[1] / D;   // 2048

    float* crow = (float*)d_ws;      // M floats
    float* q2   = crow + M;          // M floats
    float* yn   = q2 + M;            // N floats  (24 KB total)

    const int mx = (M > N ? M : N);
    qkm_precompute<<<(mx + 255) / 256, 256, 0, stream>>>(x, y, crow, q2, yn, M, N, D);

    // one wave (32 threads) per 16x16 tile; 8 waves per block
    dim3 grid(N / 16, M / (16 * 8));
    qkm_fused<<<grid, 256, 0, stream>>>(x, y, crow, q2, yn, out, N);
}